// AdaptiveFilter_71416716198124
// MI455X (gfx1250) — compile-verified
//
#include <hip/hip_runtime.h>
#include <hip/hip_bf16.h>

#define H_DIM 512
#define W_DIM 512
#define C_DIM 3
#define TILE_X 32
#define TILE_Y 16
#define HALO 3
#define TWX (TILE_X + 2 * HALO) // 38
#define TWY (TILE_Y + 2 * HALO) // 22
#define NTHREADS 256
#define LOG2E 1.44269504088896340736f

typedef __attribute__((address_space(1))) int gint_t;
typedef __attribute__((address_space(3))) int lint_t;

__device__ __forceinline__ int reflect_idx(int i, int n) {
  i = (i < 0) ? -i : i;                  // jnp reflect: -1 -> 1
  return (i >= n) ? (2 * n - 2 - i) : i; // n -> n-2
}

// CDNA5 async global->LDS copy (ASYNCcnt path). Per-lane 4B transfer.
__device__ __forceinline__ void async_g2l(const float* gp, float* lp) {
#if __has_builtin(__builtin_amdgcn_global_load_async_to_lds_b32)
  __builtin_amdgcn_global_load_async_to_lds_b32(
      (gint_t*)gp, (lint_t*)lp, /*imm offset=*/0, /*cpol=*/0);
#else
  *lp = *gp; // non-async fallback (still correct)
#endif
}

__device__ __forceinline__ void wait_async0() {
#if __has_builtin(__builtin_amdgcn_s_wait_asynccnt)
  __builtin_amdgcn_s_wait_asynccnt(0);
#else
  asm volatile("s_wait_asynccnt 0" ::: "memory");
#endif
}

__device__ __forceinline__ float fast_exp2(float x) {
#if __has_builtin(__builtin_amdgcn_exp2f)
  return __builtin_amdgcn_exp2f(x); // v_exp_f32
#else
  return exp2f(x);
#endif
}

// Load a pixel's 4x4 half-kernel logits and produce (w - max(w)) * log2e.
__device__ __forceinline__ void load_wlog(const float* wbase,
                                          float wl[4][4]) {
  const float4* wp = reinterpret_cast<const float4*>(wbase);
  const float4 r0 = wp[0], r1 = wp[1], r2 = wp[2], r3 = wp[3];
  const float a[4][4] = {{r0.x, r0.y, r0.z, r0.w},
                         {r1.x, r1.y, r1.z, r1.w},
                         {r2.x, r2.y, r2.z, r2.w},
                         {r3.x, r3.y, r3.z, r3.w}};
  float m = a[0][0];
#pragma unroll
  for (int i = 0; i < 4; ++i)
#pragma unroll
    for (int j = 0; j < 4; ++j) m = fmaxf(m, a[i][j]);
#pragma unroll
  for (int i = 0; i < 4; ++i)
#pragma unroll
    for (int j = 0; j < 4; ++j) wl[i][j] = (a[i][j] - m) * LOG2E;
}

__global__ __launch_bounds__(NTHREADS) void AdaptiveFilter_71416716198124_kernel(
    const float* __restrict__ xin, const float* __restrict__ gin,
    const float* __restrict__ w0, float* __restrict__ out) {
  __shared__ float4 gT[TWY][TWX]; // {g0,g1,g2,pad} channel-interleaved
  __shared__ float4 xT[TWY][TWX];

  const int tx = threadIdx.x, ty = threadIdx.y; // 16x16 threads
  const int bx = blockIdx.x, by = blockIdx.y, b = blockIdx.z;
  const int tid = ty * 16 + tx;

  // ---- Stage haloed guidance/x tiles into LDS via async copies ----
  float* gL = reinterpret_cast<float*>(&gT[0][0]);
  float* xL = reinterpret_cast<float*>(&xT[0][0]);
  for (int idx = tid; idx < TWY * TWX; idx += NTHREADS) {
    const int ly = idx / TWX;
    const int lx = idx - ly * TWX;
    const int gy = reflect_idx(by * TILE_Y + ly - HALO, H_DIM);
    const int gx = reflect_idx(bx * TILE_X + lx - HALO, W_DIM);
#pragma unroll
    for (int c = 0; c < C_DIM; ++c) {
      const size_t off = (((size_t)b * C_DIM + c) * H_DIM + gy) * W_DIM + gx;
      async_g2l(gin + off, gL + 4 * idx + c);
      async_g2l(xin + off, xL + 4 * idx + c);
    }
  }
  wait_async0();
  __syncthreads();

  // ---- Each thread computes TWO adjacent output pixels (xo, xo+1) ----
  const int tx2 = tx * 2;
  const int yo = by * TILE_Y + ty;
  const int xo = bx * TILE_X + tx2;
  const float* wbase =
      w0 + ((size_t)b * H_DIM * W_DIM + (size_t)yo * W_DIM + xo) * 16;

  float wlA[4][4], wlB[4][4];
  load_wlog(wbase, wlA);       // pixel A = (yo, xo)
  load_wlog(wbase + 16, wlB);  // pixel B = (yo, xo+1)

  const float4 gcA = gT[ty + HALO][tx2 + HALO];
  const float4 gcB = gT[ty + HALO][tx2 + HALO + 1];

  // out = sum(x * 2^{wl - 50*log2e*dist}) / sum(2^{wl - 50*log2e*dist});
  // the softmax denominator cancels against the final normalization.
  const float kC = -50.0f * LOG2E;
  float n0A = 0.f, n1A = 0.f, n2A = 0.f, dA = 0.f;
  float n0B = 0.f, n1B = 0.f, n2B = 0.f, dB = 0.f;

#pragma unroll
  for (int i = 0; i < 7; ++i) {
    const int mi = (i < 4) ? i : 6 - i; // reflect-expanded half-kernel row
    float4 gr[8], xr[8];                // 8 columns serve both pixels
#pragma unroll
    for (int j = 0; j < 8; ++j) {
      gr[j] = gT[ty + i][tx2 + j];
      xr[j] = xT[ty + i][tx2 + j];
    }
#pragma unroll
    for (int j = 0; j < 7; ++j) {
      const int mj = (j < 4) ? j : 6 - j;
      // pixel A: tap column j
      {
        const float s = fabsf(gr[j].x - gcA.x) + fabsf(gr[j].y - gcA.y) +
                        fabsf(gr[j].z - gcA.z);
        const float e = fast_exp2(fmaf(kC, s * s, wlA[mi][mj]));
        n0A = fmaf(e, xr[j].x, n0A);
        n1A = fmaf(e, xr[j].y, n1A);
        n2A = fmaf(e, xr[j].z, n2A);
        dA += e;
      }
      // pixel B: tap column j+1
      {
        const float s = fabsf(gr[j + 1].x - gcB.x) +
                        fabsf(gr[j + 1].y - gcB.y) +
                        fabsf(gr[j + 1].z - gcB.z);
        const float e = fast_exp2(fmaf(kC, s * s, wlB[mi][mj]));
        n0B = fmaf(e, xr[j + 1].x, n0B);
        n1B = fmaf(e, xr[j + 1].y, n1B);
        n2B = fmaf(e, xr[j + 1].z, n2B);
        dB += e;
      }
    }
  }

  const float invA = 1.0f / dA;
  const float invB = 1.0f / dB;
  const size_t hw = (size_t)H_DIM * W_DIM;
  const size_t ob = (((size_t)b * C_DIM) * H_DIM + yo) * W_DIM + xo;
  out[ob] = n0A * invA;
  out[ob + 1] = n0B * invB;
  out[ob + hw] = n1A * invA;
  out[ob + hw + 1] = n1B * invB;
  out[ob + 2 * hw] = n2A * invA;
  out[ob + 2 * hw + 1] = n2B * invB;
}

extern "C" void kernel_launch(void* const* d_in, const int* in_sizes, int n_in,
                              void* d_out, int out_size, void* d_ws,
                              size_t ws_size, hipStream_t stream) {
  (void)in_sizes; (void)n_in; (void)out_size; (void)d_ws; (void)ws_size;
  const float* x = (const float*)d_in[0];
  const float* g = (const float*)d_in[1];
  const float* w0 = (const float*)d_in[2];
  float* out = (float*)d_out;

  dim3 block(16, 16, 1);
  dim3 grid(W_DIM / TILE_X, H_DIM / TILE_Y, 2 /*batch*/);
  AdaptiveFilter_71416716198124_kernel<<<grid, block, 0, stream>>>(x, g, w0,
                                                                   out);
}